// SparseEmbeddingBag_87703232185059
// MI455X (gfx1250) — compile-verified
//
#include <hip/hip_runtime.h>

// SparseEmbeddingBag for MI455X (gfx1250).
// HBM-bound gather (~145 MB @ 23.3 TB/s => ~6.2 us floor).
// One wave32 per bag; each of the 16 rows is fetched as one fully-coalesced
// 512B global_load_async_to_lds_b128 (ASYNCcnt-tracked CDNA5 async copy),
// then consumed from LDS with ds_load_b128 + v_fmac.

#define N_PER_BAG 16
#define C_DIM     128            // floats per row
#define ROW_BYTES (C_DIM * 4)    // 512 B
#define WAVES_PER_BLOCK 8

__global__ __launch_bounds__(256) void sparse_embedding_bag_kernel(
    const int*   __restrict__ indices,   // (bags, 16)
    const float* __restrict__ scores,    // (bags, 16)
    const float* __restrict__ weight,    // (1e6, 128)
    float*       __restrict__ out,       // (bags, 128)
    int num_bags)
{
    // 8 waves * 16 rows * 128 floats = 64 KB of LDS per block
    __shared__ __align__(16) float lds[WAVES_PER_BLOCK * N_PER_BAG * C_DIM];

    const int lane          = threadIdx.x & 31;
    const int wave_in_block = threadIdx.x >> 5;
    const int bag           = blockIdx.x * WAVES_PER_BLOCK + wave_in_block;
    if (bag >= num_bags) return;   // uniform per wave -> EXEC stays all-or-nothing

    // ---- load this bag's 16 indices + 16 scores (wave-uniform addresses) ----
    int   idx[N_PER_BAG];
    float sc [N_PER_BAG];
    {
        const int4*   ip = reinterpret_cast<const int4*>  (indices + (size_t)bag * N_PER_BAG);
        const float4* sp = reinterpret_cast<const float4*>(scores  + (size_t)bag * N_PER_BAG);
#pragma unroll
        for (int q = 0; q < 4; ++q) {
            int4   iv = ip[q];
            float4 sv = sp[q];
            idx[4*q+0] = iv.x; idx[4*q+1] = iv.y; idx[4*q+2] = iv.z; idx[4*q+3] = iv.w;
            sc [4*q+0] = sv.x; sc [4*q+1] = sv.y; sc [4*q+2] = sv.z; sc [4*q+3] = sv.w;
        }
    }

    // ---- issue 16 async row gathers: global -> LDS, 512B each, coalesced ----
    // Generic pointer to LDS truncates to the wave-relative LDS byte offset.
    float* wave_lds = &lds[wave_in_block * N_PER_BAG * C_DIM];
    const unsigned lds_base = (unsigned)(uintptr_t)wave_lds;
    const unsigned long long wbase = (unsigned long long)(uintptr_t)weight;

#pragma unroll
    for (int n = 0; n < N_PER_BAG; ++n) {
        unsigned long long gaddr = wbase
                                 + (unsigned long long)(unsigned)idx[n] * ROW_BYTES
                                 + (unsigned)(lane * 16);
        unsigned laddr = lds_base + (unsigned)(n * ROW_BYTES) + (unsigned)(lane * 16);
        asm volatile("global_load_async_to_lds_b128 %0, %1, off"
                     :
                     : "v"(laddr), "v"(gaddr)
                     : "memory");
    }

    // wait for all 16 async copies to land in LDS
    asm volatile("s_wait_asynccnt 0" ::: "memory");

    // ---- weighted reduction: acc += sc[n] * row_n[lane*4 .. lane*4+3] ----
    const float4* lrow = reinterpret_cast<const float4*>(wave_lds) + lane;
    float4 acc = make_float4(0.f, 0.f, 0.f, 0.f);
#pragma unroll
    for (int n = 0; n < N_PER_BAG; ++n) {
        float4 v = lrow[n * (C_DIM / 4)];    // ds_load_b128
        acc.x += sc[n] * v.x;
        acc.y += sc[n] * v.y;
        acc.z += sc[n] * v.z;
        acc.w += sc[n] * v.w;
    }

    // ---- store 512B per bag, coalesced ----
    float4* op = reinterpret_cast<float4*>(out) + (size_t)bag * (C_DIM / 4) + lane;
    *op = acc;
}

extern "C" void kernel_launch(void* const* d_in, const int* in_sizes, int n_in,
                              void* d_out, int out_size, void* d_ws, size_t ws_size,
                              hipStream_t stream) {
    const int*   indices = (const int*)  d_in[0];   // (B,S,N) int32
    const float* scores  = (const float*)d_in[1];   // (B,S,N) f32
    const float* weight  = (const float*)d_in[2];   // (NE,C)  f32
    float*       out     = (float*)d_out;           // (B,S,C) f32

    const int num_bags = in_sizes[0] / N_PER_BAG;   // B*S = 16384
    const int blocks   = (num_bags + WAVES_PER_BLOCK - 1) / WAVES_PER_BLOCK;

    sparse_embedding_bag_kernel<<<blocks, 32 * WAVES_PER_BLOCK, 0, stream>>>(
        indices, scores, weight, out, num_bags);
}